// SpatialSatCrossAttention_85968065397290
// MI455X (gfx1250) — compile-verified
//
#include <hip/hip_runtime.h>
#include <hip/hip_bf16.h>
#include <math.h>

// ---------------- problem constants (from reference) ----------------
#define NCAM   6
#define NQ     10000
#define DIMS   256
#define NH     8
#define DH     32      // DIMS / NH
#define NLVL   4
#define NPTS   8
#define DHGT   4       // D_HEIGHT
#define SUMHW  5440    // 64*64 + 32*32 + 16*16 + 8*8

typedef float v2f __attribute__((ext_vector_type(2)));
typedef float v8f __attribute__((ext_vector_type(8)));

// =====================================================================
// Generic fp32 WMMA GEMM:  C[M,N] = A[M,K] @ B[K,N] + bias[N] (+ residual)
// One wave (32 threads) per 16x16 output tile, V_WMMA_F32_16X16X4_F32.
//   A fragment (16x4):  lane m = lane&15, kk = (lane>=16)?2:0 -> {A[m][k+kk], A[m][k+kk+1]}
//   B fragment (4x16):  lane n = lane&15, kk likewise          -> {B[k+kk][n], B[k+kk+1][n]}
//   C/D (16x16, 8 VGPR): vgpr r, lanes 0-15 -> (M=r, N=lane); lanes 16-31 -> (M=r+8, N=lane-16)
// =====================================================================
__global__ void ssca_gemm16_wmma_f32(const float* __restrict__ A,
                                     const float* __restrict__ B,
                                     const float* __restrict__ bias,
                                     const float* __restrict__ residual,
                                     float* __restrict__ C,
                                     int M, int N, int K) {
    const int ntiles = N >> 4;
    const int tile   = blockIdx.x;
    const int mt     = tile / ntiles;
    const int nt     = tile - mt * ntiles;
    const int m0     = mt << 4;
    const int n0     = nt << 4;

    const int lane = threadIdx.x & 31;  // wave32
    const int r16  = lane & 15;
    const int kk   = (lane >> 4) << 1;  // 0 or 2

    const float* __restrict__ arow = A + (size_t)(m0 + r16) * K;
    const float* __restrict__ bcol = B + n0 + r16;

    v8f acc = {};
#pragma unroll 4
    for (int k = 0; k < K; k += 4) {
        v2f a, b;
        a.x = arow[k + kk];
        a.y = arow[k + kk + 1];
        b.x = bcol[(size_t)(k + kk) * N];
        b.y = bcol[(size_t)(k + kk + 1) * N];
        acc = __builtin_amdgcn_wmma_f32_16x16x4_f32(
            /*neg_a=*/false, a, /*neg_b=*/false, b,
            /*c_mod=*/(short)0, acc, /*reuse_a=*/false, /*reuse_b=*/false);
    }

    const int col   = n0 + r16;
    const float bv  = bias ? bias[col] : 0.0f;
    const int rbase = (lane < 16) ? 0 : 8;
#pragma unroll
    for (int r = 0; r < 8; ++r) {
        const int row = m0 + rbase + r;
        float v = acc[r] + bv;
        if (residual) v += residual[(size_t)row * N + col];
        C[(size_t)row * N + col] = v;
    }
}

// =====================================================================
// Softmax over Lv*P = 32 logits per (q, head).  Block = 1 query (8 waves),
// each wave32 handles one head; pure lane-shuffle reduction (wave32).
// =====================================================================
__global__ void ssca_softmax32(float* __restrict__ w) {
    const int q = blockIdx.x;
    const int t = threadIdx.x;  // 0..255, wave boundary == head boundary
    const size_t idx = (size_t)q * 256 + t;
    float v = w[idx];

    float m = v;
#pragma unroll
    for (int s = 16; s > 0; s >>= 1) m = fmaxf(m, __shfl_xor(m, s, 32));
    float e = __expf(v - m);
    float sum = e;
#pragma unroll
    for (int s = 16; s > 0; s >>= 1) sum += __shfl_xor(sum, s, 32);
    w[idx] = e / sum;
}

// =====================================================================
// Deformable sampler + camera mask reduction.
// Block = 1 query (256 threads = 8 waves).  wave = head, lane = channel.
// Gathers hit the projected value tensor (33 MB -> resident in 192 MB L2);
// every corner read is a coalesced 128B burst (32 lanes x 4B contiguous).
// =====================================================================
__global__ void ssca_sampler(const float* __restrict__ Vp,     // [NCAM*SUMHW][DIMS]
                             const float* __restrict__ offr,   // [NQ][512] raw offsets
                             const float* __restrict__ attw,   // [NQ][256] softmaxed
                             const float* __restrict__ ref,    // [NCAM][NQ][DHGT][2]
                             const int*   __restrict__ vox,    // [NCAM][NQ][DHGT]
                             float* __restrict__ slots)        // [NQ][DIMS]
{
    __shared__ float s_off[NH * NLVL * NPTS * 2];  // 512
    __shared__ float s_atw[NH * NLVL * NPTS];      // 256

    const int q = blockIdx.x;
    const int t = threadIdx.x;
    s_off[t]       = offr[(size_t)q * 512 + t];
    s_off[t + 256] = offr[(size_t)q * 512 + 256 + t];
    s_atw[t]       = attw[(size_t)q * 256 + t];
    __syncthreads();

    const int head = t >> 5;
    const int ch   = t & 31;

    const int   LW[NLVL]    = {64, 32, 16, 8};
    const int   LSTART[NLVL]= {0, 4096, 5120, 5376};
    const float LINV[NLVL]  = {1.0f / 64.0f, 1.0f / 32.0f, 1.0f / 16.0f, 1.0f / 8.0f};

    float acc = 0.0f;
    float cnt = 0.0f;

    for (int cam = 0; cam < NCAM; ++cam) {
        const int* vm = vox + ((size_t)cam * NQ + q) * DHGT;
        const int valid = vm[0] | vm[1] | vm[2] | vm[3];
        if (!valid) continue;
        cnt += 1.0f;

        const float* refq = ref + ((size_t)cam * NQ + q) * (DHGT * 2);
        float rx[DHGT], ry[DHGT];
#pragma unroll
        for (int d = 0; d < DHGT; ++d) { rx[d] = refq[2 * d]; ry[d] = refq[2 * d + 1]; }

#pragma unroll
        for (int lvl = 0; lvl < NLVL; ++lvl) {
            const int   w   = LW[lvl];
            const int   h   = LW[lvl];             // square levels
            const float inv = LINV[lvl];           // 1/w == 1/h
            const float* vb = Vp + ((size_t)(cam * SUMHW + LSTART[lvl])) * DIMS
                                 + head * DH + ch;
#pragma unroll
            for (int p = 0; p < NPTS; ++p) {
                const int d = p & 3;               // p = (p/4)*DHGT + d
                const float lx = rx[d] + s_off[head * 64 + lvl * 16 + p * 2 + 0] * inv;
                const float ly = ry[d] + s_off[head * 64 + lvl * 16 + p * 2 + 1] * inv;
                const float x  = lx * (float)w - 0.5f;
                const float y  = ly * (float)h - 0.5f;
                const float xf = floorf(x);
                const float yf = floorf(y);
                const int   x0 = (int)xf;
                const int   y0 = (int)yf;
                const float fx = x - xf;
                const float fy = y - yf;

                float pacc = 0.0f;
                {   // (x0, y0)
                    if (x0 >= 0 && x0 < w && y0 >= 0 && y0 < h)
                        pacc += (1.0f - fx) * (1.0f - fy) * vb[(size_t)(y0 * w + x0) * DIMS];
                }
                {   // (x0+1, y0)
                    const int xi = x0 + 1;
                    if (xi >= 0 && xi < w && y0 >= 0 && y0 < h)
                        pacc += fx * (1.0f - fy) * vb[(size_t)(y0 * w + xi) * DIMS];
                }
                {   // (x0, y0+1)
                    const int yi = y0 + 1;
                    if (x0 >= 0 && x0 < w && yi >= 0 && yi < h)
                        pacc += (1.0f - fx) * fy * vb[(size_t)(yi * w + x0) * DIMS];
                }
                {   // (x0+1, y0+1)
                    const int xi = x0 + 1, yi = y0 + 1;
                    if (xi >= 0 && xi < w && yi >= 0 && yi < h)
                        pacc += fx * fy * vb[(size_t)(yi * w + xi) * DIMS];
                }
                acc += s_atw[head * 32 + lvl * 8 + p] * pacc;
            }
        }
    }

    cnt = fmaxf(cnt, 1.0f);
    slots[(size_t)q * DIMS + head * DH + ch] = acc / cnt;
}

// =====================================================================
extern "C" void kernel_launch(void* const* d_in, const int* in_sizes, int n_in,
                              void* d_out, int out_size, void* d_ws, size_t ws_size,
                              hipStream_t stream) {
    const float* query   = (const float*)d_in[0];
    // d_in[1] = key : unused by the deformable-attention math
    const float* value   = (const float*)d_in[2];   // [NCAM,SUMHW,1,DIMS] == [32640][256]
    const float* ref     = (const float*)d_in[3];   // [NCAM,1,NQ,DHGT,2]
    const int*   vox     = (const int*)  d_in[4];   // [NCAM,1,NQ,DHGT]
    // d_in[5] spatial_shapes / d_in[6] level_start_index : compile-time constants
    const float* value_W = (const float*)d_in[7];
    const float* value_b = (const float*)d_in[8];
    const float* off_W   = (const float*)d_in[9];
    const float* off_b   = (const float*)d_in[10];
    const float* attw_W  = (const float*)d_in[11];
    const float* attw_b  = (const float*)d_in[12];
    const float* out_W   = (const float*)d_in[13];
    const float* out_b   = (const float*)d_in[14];

    float* out = (float*)d_out;

    // workspace layout (floats): Vp | off_raw | attw | slots_pre  (~74.4 MB)
    float* ws    = (float*)d_ws;
    float* Vp    = ws;                                   // NCAM*SUMHW*DIMS = 8,355,840
    float* offr  = Vp   + (size_t)NCAM * SUMHW * DIMS;   // NQ*512 = 5,120,000
    float* atw   = offr + (size_t)NQ * 512;              // NQ*256 = 2,560,000
    float* slots = atw  + (size_t)NQ * 256;              // NQ*256 = 2,560,000

    // 1) value projection: [32640,256] = value @ value_W + value_b
    ssca_gemm16_wmma_f32<<<(NCAM * SUMHW / 16) * (DIMS / 16), 32, 0, stream>>>(
        value, value_W, value_b, nullptr, Vp, NCAM * SUMHW, DIMS, DIMS);

    // 2) sampling offsets: [10000,512] = query @ off_W + off_b
    ssca_gemm16_wmma_f32<<<(NQ / 16) * (512 / 16), 32, 0, stream>>>(
        query, off_W, off_b, nullptr, offr, NQ, 512, DIMS);

    // 3) attention logits: [10000,256] = query @ attw_W + attw_b
    ssca_gemm16_wmma_f32<<<(NQ / 16) * (256 / 16), 32, 0, stream>>>(
        query, attw_W, attw_b, nullptr, atw, NQ, 256, DIMS);

    // 4) softmax over the 32 (level,point) logits per (q, head)
    ssca_softmax32<<<NQ, 256, 0, stream>>>(atw);

    // 5) deformable bilinear sampling + camera mask reduction
    ssca_sampler<<<NQ, 256, 0, stream>>>(Vp, offr, atw, ref, vox, slots);

    // 6) output projection + residual: out = slots @ out_W + out_b + query
    ssca_gemm16_wmma_f32<<<(NQ / 16) * (256 / 16), 32, 0, stream>>>(
        slots, out_W, out_b, query, out, NQ, 256, DIMS);
}